// Attention_core_41343355191617
// MI455X (gfx1250) — compile-verified
//
#include <hip/hip_runtime.h>
#include <cstdint>

typedef __attribute__((ext_vector_type(2))) float v2f;
typedef __attribute__((ext_vector_type(8))) float v8f;

#define NB 16
#define NH 32
#define SEQ 4096
#define DH 128
#define NWAVES 8
#define TILE 16
#define KSTRIDE 132                 // padded K-tile row stride (floats): bank-conflict-free ds_load_b64
#define KTILEF (TILE * KSTRIDE)     // 2112 floats per tile buffer

// dynamic smem layout (float indices)
#define SM_Q    0
#define SM_ACC  128                       // 8 waves * 128
#define SM_M    (SM_ACC + NWAVES * DH)    // 1152
#define SM_L    (SM_M + NWAVES)           // 1160
#define SM_K    1184                      // 16B aligned
#define SM_TOTALF (SM_K + NWAVES * 2 * KTILEF)
#define SMEM_BYTES (SM_TOTALF * 4)

__global__ __launch_bounds__(256) void attn_decode_f32(
    const float* __restrict__ q, const float* __restrict__ kc,
    const float* __restrict__ vc, const int* __restrict__ ntok,
    float* __restrict__ out)
{
    extern __shared__ float smem[];
    const int L   = ntok[0] + 1;
    const int bh  = blockIdx.x;
    const int tid = threadIdx.x;
    const int wave = tid >> 5;
    const int lane = tid & 31;

    const float* qg = q  + (size_t)bh * DH;
    const float* kb = kc + (size_t)bh * SEQ * DH;
    const float* vb = vc + (size_t)bh * SEQ * DH;

    if (tid < DH) smem[SM_Q + tid] = qg[tid];
    __syncthreads();

    // Hoist q as WMMA B-fragments (4x16, q broadcast across N):
    // lane half h = lane>>4 supplies K rows {2h, 2h+1} of each 4-chunk.
    const int ph = (lane >> 4) << 1;
    v2f qf[32];
#pragma unroll
    for (int c = 0; c < 32; ++c) {
        qf[c].x = smem[SM_Q + 4 * c + ph];
        qf[c].y = smem[SM_Q + 4 * c + ph + 1];
    }

    const int T = (L + TILE - 1) / TILE;
    float* mykt = &smem[SM_K + wave * 2 * KTILEF];
    const unsigned lds_base = (unsigned)(uintptr_t)mykt;   // low 32 bits of generic addr = LDS offset

    const int arow = lane & 15;   // A-fragment: key row
    const int dl   = lane << 2;   // this lane's d-slice (4 floats) for V/out

    float m_run = -3.0e38f, l_run = 0.0f;
    float acc0 = 0.f, acc1 = 0.f, acc2 = 0.f, acc3 = 0.f;

    auto issue = [&](int tt, int buf) {
#pragma unroll
        for (int r = 0; r < TILE; ++r) {
            int key = tt * TILE + r;
            if (key >= L) key = L - 1;                       // clamp tail (masked later)
            const float* g = kb + (size_t)key * DH + dl;     // coalesced: 32 lanes x 16B
            unsigned ldsa = lds_base + (unsigned)(buf * KTILEF + r * KSTRIDE) * 4u
                                     + (unsigned)(lane << 4);
            unsigned long long ga = (unsigned long long)(uintptr_t)g;
            asm volatile("global_load_async_to_lds_b128 %0, %1, off"
                         :: "v"(ldsa), "v"(ga) : "memory");
        }
    };

    int tt = wave;
    if (tt < T) issue(tt, 0);
    int buf = 0;
    for (; tt < T; tt += NWAVES) {
        const int nx = tt + NWAVES;
        if (nx < T) {
            issue(nx, buf ^ 1);                               // prefetch next tile
            asm volatile("s_wait_asynccnt 0x10" ::: "memory"); // oldest 16 (current tile) done
        } else {
            asm volatile("s_wait_asynccnt 0x0" ::: "memory");
        }

        // ---- scores: 16 keys x q, via V_WMMA_F32_16X16X4_F32, 32 chunks of K=4 ----
        const float* abase = mykt + buf * KTILEF + arow * KSTRIDE + ph;
        v8f c = {};
#pragma unroll
        for (int ch = 0; ch < 32; ++ch) {
            v2f a = *(const v2f*)(abase + 4 * ch);            // conflict-free ds_load_b64
            c = __builtin_amdgcn_wmma_f32_16x16x4_f32(
                    false, a, false, qf[ch], (short)0, c, false, false);
        }

        // extract scores (C row m replicated across N): broadcast via v_readlane
        float s[16];
#pragma unroll
        for (int j = 0; j < 8; ++j) {
            s[j]     = __int_as_float(__builtin_amdgcn_readlane(__float_as_int(c[j]), 0));
            s[j + 8] = __int_as_float(__builtin_amdgcn_readlane(__float_as_int(c[j]), 16));
        }

        // ---- online softmax + PV (V rows loaded coalesced b128, lane owns 4 dims) ----
        const int kb0 = tt * TILE;
        float tmax = -3.0e38f;
#pragma unroll
        for (int j = 0; j < 16; ++j)
            if (kb0 + j < L) tmax = fmaxf(tmax, s[j]);
        const float m_new = fmaxf(m_run, tmax);
        const float corr = __expf(m_run - m_new);
        l_run *= corr; acc0 *= corr; acc1 *= corr; acc2 *= corr; acc3 *= corr;
#pragma unroll
        for (int j = 0; j < 16; ++j) {
            const int key = kb0 + j;
            if (key < L) {
                const float p = __expf(s[j] - m_new);
                l_run += p;
                const float4 v4 = *(const float4*)(vb + (size_t)key * DH + dl);
                acc0 = fmaf(p, v4.x, acc0);
                acc1 = fmaf(p, v4.y, acc1);
                acc2 = fmaf(p, v4.z, acc2);
                acc3 = fmaf(p, v4.w, acc3);
            }
        }
        m_run = m_new;
        buf ^= 1;
    }

    // ---- cross-wave merge ----
    smem[SM_M + wave] = m_run;
    smem[SM_L + wave] = l_run;
    float* acw = &smem[SM_ACC + wave * DH + dl];
    acw[0] = acc0; acw[1] = acc1; acw[2] = acc2; acw[3] = acc3;
    __syncthreads();

    if (tid < DH) {
        float M = -3.0e38f;
#pragma unroll
        for (int w = 0; w < NWAVES; ++w) M = fmaxf(M, smem[SM_M + w]);
        float lt = 0.f, o = 0.f;
#pragma unroll
        for (int w = 0; w < NWAVES; ++w) {
            const float sc = __expf(smem[SM_M + w] - M);
            lt += smem[SM_L + w] * sc;
            o  += smem[SM_ACC + w * DH + tid] * sc;
        }
        out[(size_t)bh * DH + tid] = o / lt;
    }
}

extern "C" void kernel_launch(void* const* d_in, const int* in_sizes, int n_in,
                              void* d_out, int out_size, void* d_ws, size_t ws_size,
                              hipStream_t stream) {
    const float* q  = (const float*)d_in[0];
    const float* k  = (const float*)d_in[1];
    const float* v  = (const float*)d_in[2];
    const int*   nt = (const int*)d_in[3];
    float* out = (float*)d_out;

    dim3 grid(NB * NH);   // 512 blocks: one per (b,h)
    dim3 block(256);      // 8 wave32
    attn_decode_f32<<<grid, block, SMEM_BYTES, stream>>>(q, k, v, nt, out);
}